// AttentionGate_13262859010321
// MI455X (gfx1250) — compile-verified
//
#include <hip/hip_runtime.h>
#include <math.h>

// ---------------------------------------------------------------------------
// AttentionGate on MI455X (gfx1250): fp32 WMMA 16x16x4 dual-GEMM + 3-pass
// GroupNorm reductions. Memory-bound design: only the 1-channel psi
// pre-activation (2MB) is materialized; the 64-channel conv outputs are
// recomputed in pass 2 instead of round-tripping 536MB through HBM.
// ---------------------------------------------------------------------------

#define NSP   262144      // 64*64*64 spatial points per batch
#define CGIN  128         // g input channels
#define CXIN  64          // x input channels
#define CI    64          // intermediate channels
#define NB    2           // batch
#define EPSV  1e-5f
#define SLOPE 0.01f
#define TPW   8           // 16-point tiles per wave per block

typedef float v2f __attribute__((ext_vector_type(2)));
typedef float v8f __attribute__((ext_vector_type(8)));

#define WG_LDSTRIDE 130   // 128 + 2 pad (keeps 8B align, kills bank conflicts)
#define WX_LDSTRIDE 66    // 64 + 2 pad

__device__ __forceinline__ float half_reduce(float v) {
  // reduces lanes 0..15 and 16..31 independently (masks never cross halves)
  v += __shfl_xor(v, 1, 32);
  v += __shfl_xor(v, 2, 32);
  v += __shfl_xor(v, 4, 32);
  v += __shfl_xor(v, 8, 32);
  return v;
}

// Compute yg = Wg * g  (64x128 @ 128x16) and yx = Wx * x (64x64 @ 64x16)
// for one 16-point tile. A frag: M = lane&15, K = 2*(lane>=16)+r (float2 from
// LDS). B frag: K = 2*(lane>=16)+r, N = lane&15 (channel-major global loads).
// D acc[ot]: channel = ot*16 + 8*(lane>=16) + r, point = lane&15.
__device__ __forceinline__ void dual_gemm_tile(const float* __restrict__ gb,
                                               const float* __restrict__ xb,
                                               const float* __restrict__ sWg,
                                               const float* __restrict__ sWx,
                                               int sp, int m, int hi,
                                               v8f accg[4], v8f accx[4]) {
  const int kh = 2 * hi;
#pragma unroll 4
  for (int k0 = 0; k0 < CGIN; k0 += 4) {
    const float* p = gb + (size_t)(k0 + kh) * NSP + sp;
    v2f bf; bf.x = p[0]; bf.y = p[NSP];
#pragma unroll
    for (int ot = 0; ot < 4; ++ot) {
      v2f af = *(const v2f*)(sWg + (ot * 16 + m) * WG_LDSTRIDE + k0 + kh);
      accg[ot] = __builtin_amdgcn_wmma_f32_16x16x4_f32(
          false, af, false, bf, (short)0, accg[ot], false, false);
    }
  }
#pragma unroll 4
  for (int k0 = 0; k0 < CXIN; k0 += 4) {
    const float* p = xb + (size_t)(k0 + kh) * NSP + sp;
    v2f bf; bf.x = p[0]; bf.y = p[NSP];
#pragma unroll
    for (int ot = 0; ot < 4; ++ot) {
      v2f af = *(const v2f*)(sWx + (ot * 16 + m) * WX_LDSTRIDE + k0 + kh);
      accx[ot] = __builtin_amdgcn_wmma_f32_16x16x4_f32(
          false, af, false, bf, (short)0, accx[ot], false, false);
    }
  }
}

// ---------------------------------------------------------------------------
__global__ void init_stats(float* stats, int n) {
  int i = blockIdx.x * blockDim.x + threadIdx.x;
  if (i < n) stats[i] = 0.0f;
}

// Pass 1: dual GEMM, accumulate per-(batch, group) sum / sumsq for yg and yx.
__global__ __launch_bounds__(256) void pass1_stats(
    const float* __restrict__ g, const float* __restrict__ x,
    const float* __restrict__ Wg, const float* __restrict__ Wx,
    float* __restrict__ statsG, float* __restrict__ statsX) {
  __shared__ float sWg[CI * WG_LDSTRIDE];
  __shared__ float sWx[CI * WX_LDSTRIDE];
  for (int i = threadIdx.x; i < CI * CGIN; i += 256)
    sWg[(i >> 7) * WG_LDSTRIDE + (i & 127)] = Wg[i];
  for (int i = threadIdx.x; i < CI * CXIN; i += 256)
    sWx[(i >> 6) * WX_LDSTRIDE + (i & 63)] = Wx[i];
  __syncthreads();

  const int lane = threadIdx.x & 31, wave = threadIdx.x >> 5;
  const int m = lane & 15, hi = lane >> 4;
  const int bpb = NSP / (8 * TPW * 16);              // blocks per batch = 256
  const int b = blockIdx.x / bpb;
  const int sbase = (blockIdx.x % bpb) * (8 * TPW * 16) + wave * (TPW * 16);
  const float* gb = g + (size_t)b * CGIN * NSP;
  const float* xb = x + (size_t)b * CXIN * NSP;

  float sg[4] = {0, 0, 0, 0}, qg[4] = {0, 0, 0, 0};
  float sx[4] = {0, 0, 0, 0}, qx[4] = {0, 0, 0, 0};

  for (int t = 0; t < TPW; ++t) {
    const int sp = sbase + t * 16 + m;
    v8f accg[4] = {}; v8f accx[4] = {};
    dual_gemm_tile(gb, xb, sWg, sWx, sp, m, hi, accg, accx);
#pragma unroll
    for (int ot = 0; ot < 4; ++ot)
#pragma unroll
      for (int r = 0; r < 8; ++r) {
        float v = accg[ot][r]; sg[ot] += v; qg[ot] += v * v;
        v = accx[ot][r];       sx[ot] += v; qx[ot] += v * v;
      }
  }
#pragma unroll
  for (int ot = 0; ot < 4; ++ot) {
    sg[ot] = half_reduce(sg[ot]); qg[ot] = half_reduce(qg[ot]);
    sx[ot] = half_reduce(sx[ot]); qx[ot] = half_reduce(qx[ot]);
  }
  if (m == 0) {   // lanes 0 and 16: one per lane-half (= one per group pair)
#pragma unroll
    for (int ot = 0; ot < 4; ++ot) {
      const int grp = (b * 8 + 2 * ot + hi) * 2;
      atomicAdd(&statsG[grp], sg[ot]); atomicAdd(&statsG[grp + 1], qg[ot]);
      atomicAdd(&statsX[grp], sx[ot]); atomicAdd(&statsX[grp + 1], qx[ot]);
    }
  }
}

// Pass 2: recompute GEMM, apply GroupNorm + LeakyReLU, psi dot product,
// store 1-channel s, accumulate per-batch psi sum / sumsq.
__global__ __launch_bounds__(256) void pass2_psi(
    const float* __restrict__ g, const float* __restrict__ x,
    const float* __restrict__ Wg, const float* __restrict__ Wx,
    const float* __restrict__ Ggw, const float* __restrict__ Ggb,
    const float* __restrict__ Xgw, const float* __restrict__ Xgb,
    const float* __restrict__ Pw,
    const float* __restrict__ statsG, const float* __restrict__ statsX,
    float* __restrict__ sbuf, float* __restrict__ statsP) {
  __shared__ float sWg[CI * WG_LDSTRIDE];
  __shared__ float sWx[CI * WX_LDSTRIDE];
  __shared__ float sGw[CI], sGb[CI], sXw[CI], sXb[CI], sPw[CI];
  for (int i = threadIdx.x; i < CI * CGIN; i += 256)
    sWg[(i >> 7) * WG_LDSTRIDE + (i & 127)] = Wg[i];
  for (int i = threadIdx.x; i < CI * CXIN; i += 256)
    sWx[(i >> 6) * WX_LDSTRIDE + (i & 63)] = Wx[i];
  if (threadIdx.x < CI) {
    sGw[threadIdx.x] = Ggw[threadIdx.x]; sGb[threadIdx.x] = Ggb[threadIdx.x];
    sXw[threadIdx.x] = Xgw[threadIdx.x]; sXb[threadIdx.x] = Xgb[threadIdx.x];
    sPw[threadIdx.x] = Pw[threadIdx.x];
  }
  __syncthreads();

  const int lane = threadIdx.x & 31, wave = threadIdx.x >> 5;
  const int m = lane & 15, hi = lane >> 4;
  const int bpb = NSP / (8 * TPW * 16);
  const int b = blockIdx.x / bpb;
  const int sbase = (blockIdx.x % bpb) * (8 * TPW * 16) + wave * (TPW * 16);
  const float* gb = g + (size_t)b * CGIN * NSP;
  const float* xb = x + (size_t)b * CXIN * NSP;

  // per-group mean / rstd for this lane's groups (group = 2*ot + hi)
  const float cntG = 8.0f * (float)NSP;
  float mg[4], rg[4], mx[4], rx[4];
#pragma unroll
  for (int ot = 0; ot < 4; ++ot) {
    const int grp = (b * 8 + 2 * ot + hi) * 2;
    float s0 = statsG[grp], q0 = statsG[grp + 1];
    float mean = s0 / cntG;
    mg[ot] = mean;
    rg[ot] = rsqrtf(fmaxf(q0 / cntG - mean * mean, 0.0f) + EPSV);
    s0 = statsX[grp]; q0 = statsX[grp + 1];
    mean = s0 / cntG;
    mx[ot] = mean;
    rx[ot] = rsqrtf(fmaxf(q0 / cntG - mean * mean, 0.0f) + EPSV);
  }

  float psum = 0.0f, pq = 0.0f;
  for (int t = 0; t < TPW; ++t) {
    const int sp = sbase + t * 16 + m;
    v8f accg[4] = {}; v8f accx[4] = {};
    dual_gemm_tile(gb, xb, sWg, sWx, sp, m, hi, accg, accx);

    float part = 0.0f;
#pragma unroll
    for (int ot = 0; ot < 4; ++ot)
#pragma unroll
      for (int r = 0; r < 8; ++r) {
        const int c = ot * 16 + 8 * hi + r;
        float g1 = (accg[ot][r] - mg[ot]) * rg[ot] * sGw[c] + sGb[c];
        float x1 = (accx[ot][r] - mx[ot]) * rx[ot] * sXw[c] + sXb[c];
        float tt = g1 + x1;
        float a = tt >= 0.0f ? tt : SLOPE * tt;   // LeakyReLU(0.01)
        part += sPw[c] * a;
      }
    // combine the two lane-halves: full 64-channel dot product per point
    float sv = part + __shfl_xor(part, 16, 32);
    if (hi == 0) {
      sbuf[(size_t)b * NSP + sp] = sv;
      psum += sv; pq += sv * sv;
    }
  }
  psum = half_reduce(psum); pq = half_reduce(pq);
  if (lane == 0) {
    atomicAdd(&statsP[b * 2], psum);
    atomicAdd(&statsP[b * 2 + 1], pq);
  }
}

// Pass 3: out = x * sigmoid(GN_1(s)); one thread per spatial point, loops
// the 64 x-channels (stride NSP -> fully coalesced per iteration).
__global__ __launch_bounds__(256) void pass3_out(
    const float* __restrict__ x, const float* __restrict__ sbuf,
    const float* __restrict__ statsP, const float* __restrict__ pgw,
    const float* __restrict__ pgb, float* __restrict__ out) {
  const int idx = blockIdx.x * 256 + threadIdx.x;   // 0 .. NB*NSP-1
  const int b = idx >> 18;                          // / NSP
  const int sp = idx & (NSP - 1);
  const float mean = statsP[b * 2] / (float)NSP;
  const float var = statsP[b * 2 + 1] / (float)NSP - mean * mean;
  const float z = (sbuf[idx] - mean) * rsqrtf(fmaxf(var, 0.0f) + EPSV) * pgw[0] + pgb[0];
  const float psi = 1.0f / (1.0f + __expf(-z));
  const float* xb = x + (size_t)b * CXIN * NSP + sp;
  float* ob = out + (size_t)b * CXIN * NSP + sp;
#pragma unroll 4
  for (int c = 0; c < CXIN; ++c)
    ob[(size_t)c * NSP] = xb[(size_t)c * NSP] * psi;
  (void)sp;
}

// ---------------------------------------------------------------------------
extern "C" void kernel_launch(void* const* d_in, const int* in_sizes, int n_in,
                              void* d_out, int out_size, void* d_ws, size_t ws_size,
                              hipStream_t stream) {
  (void)in_sizes; (void)n_in; (void)out_size; (void)ws_size;
  const float* g   = (const float*)d_in[0];
  const float* x   = (const float*)d_in[1];
  const float* Wg  = (const float*)d_in[2];
  const float* Ggw = (const float*)d_in[3];
  const float* Ggb = (const float*)d_in[4];
  const float* Wx  = (const float*)d_in[5];
  const float* Xgw = (const float*)d_in[6];
  const float* Xgb = (const float*)d_in[7];
  const float* Pw  = (const float*)d_in[8];
  const float* Pgw = (const float*)d_in[9];
  const float* Pgb = (const float*)d_in[10];
  float* out = (float*)d_out;

  // workspace: psi pre-activation (NB*NSP floats) + stat scratch (68 floats)
  float* sbuf   = (float*)d_ws;
  float* statsG = sbuf + (size_t)NB * NSP;  // [2][8][2]
  float* statsX = statsG + 32;              // [2][8][2]
  float* statsP = statsX + 32;              // [2][2]

  init_stats<<<1, 128, 0, stream>>>(statsG, 68);

  const int bpb = NSP / (8 * TPW * 16);     // 256 blocks per batch
  dim3 gemm_grid(NB * bpb);                 // 512
  pass1_stats<<<gemm_grid, 256, 0, stream>>>(g, x, Wg, Wx, statsG, statsX);
  pass2_psi<<<gemm_grid, 256, 0, stream>>>(g, x, Wg, Wx, Ggw, Ggb, Xgw, Xgb,
                                           Pw, statsG, statsX, sbuf, statsP);
  pass3_out<<<(NB * NSP) / 256, 256, 0, stream>>>(x, sbuf, statsP, Pgw, Pgb, out);
}